// FMFMNeuron_34033320854100
// MI455X (gfx1250) — compile-verified
//
#include <hip/hip_runtime.h>
#include <cstdint>

// LIF (snntorch Leaky, reset='subtract') over T=50 steps, B=262144 neurons,
// 2 inputs/neuron. Pure bandwidth problem (~157 MB moved, ~79 MFLOP), so we
// optimize the data path: async global->LDS double-buffered staging
// (ASYNCcnt) + wave32-coalesced b128 loads and b32 stores.

#define T_STEPS   50
#define BETA      0.9f
#define THRESH    1.0f
#define BLOCK     256

__global__ __launch_bounds__(BLOCK) void lif_async_kernel(
    const float* __restrict__ x,   // [T, B, 2]
    const float* __restrict__ W,   // [1, 2]
    float* __restrict__ out,       // [T, B, 1]
    int B)
{
    // Two 4KB stage buffers; each stage = 2 timesteps x 256 neurons x 2 floats.
    // This is the ONLY __shared__ object, so its static-LDS byte offset is 0:
    // the async-copy destinations are plain integer LDS byte addresses.
    __shared__ float buf[2][2 * BLOCK * 2];   // [stage][tsub*512 + n*2 + i]

    const int tid = threadIdx.x;
    const int b0  = blockIdx.x * BLOCK;
    const int b   = b0 + tid;

    const float w0 = W[0];
    const float w1 = W[1];

    // Async-copy addressing: 256 lanes x 16B = 4KB = 2 timesteps of this
    // block's 2KB input row. Lane i fetches chunk j=i&127 of timestep tsub=i>>7.
    const uint32_t tsub = (uint32_t)(tid >> 7);
    const uint32_t j    = (uint32_t)(tid & 127);
    // byte offset into x for the stage starting at timestep t:
    //   ((t + tsub)*B + b0 + 2*j) * 8
    uint32_t voff = (tsub * (uint32_t)B + (uint32_t)b0 + 2u * j) * 8u;
    const uint32_t voff_step = 2u * (uint32_t)B * 8u;   // advance 2 timesteps

    // Flat pointer to the LDS buffer, passed (unused) into each async asm so
    // that `buf` ESCAPES: the "memory" clobber then forces the compiler to
    // assume the asm writes buf, keeping the ds_load consumers + allocation.
    float* lds_escape = &buf[0][0];

    // Prologue: stage timesteps {0,1} into buffer 0.
    {
        uint32_t dst = (uint32_t)(tid * 16);            // buffer 0 base = 0
        asm volatile("global_load_async_to_lds_b128 %0, %1, %2 offset:0"
                     :: "v"(dst), "v"(voff), "s"(x), "v"(lds_escape)
                     : "memory");
        voff += voff_step;
    }
    asm volatile("s_wait_asynccnt 0x0" ::: "memory");
    __syncthreads();

    float mem = 0.0f;
    int sb = 0;

    for (int tt = 0; tt < T_STEPS; tt += 2) {
        // Overlap: issue the async fetch of the NEXT stage into the other
        // buffer, then compute the CURRENT stage from LDS.
        if (tt + 2 < T_STEPS) {
            uint32_t dst = (uint32_t)((sb ^ 1) * 4096 + tid * 16);
            asm volatile("global_load_async_to_lds_b128 %0, %1, %2 offset:0"
                         :: "v"(dst), "v"(voff), "s"(x), "v"(lds_escape)
                         : "memory");
            voff += voff_step;
        }

        // Two LIF steps from the current stage buffer.
        // Lane L reads 8B at LDS byte offset 8L (+2KB): banks 2L,2L+1 -> no conflicts.
        #pragma unroll
        for (int k = 0; k < 2; ++k) {
            const float2 xv =
                *(const float2*)&buf[sb][(unsigned)k * 512u + 2u * (unsigned)tid];
            float cur   = fmaf(xv.x, w0, xv.y * w1);
            float reset = (mem > THRESH) ? THRESH : 0.0f;     // H(mem_{t-1}-1)*thr
            mem         = fmaf(BETA, mem, cur - reset);
            float spk   = (mem > THRESH) ? 1.0f : 0.0f;       // H(mem_t - 1)
            out[(tt + k) * B + b] = spk;
        }

        // Next stage's async writes must be LDS-visible to the whole block
        // before anyone reads them: per-wave ASYNCcnt drain, then barrier.
        asm volatile("s_wait_asynccnt 0x0" ::: "memory");
        __syncthreads();
        sb ^= 1;
    }
}

extern "C" void kernel_launch(void* const* d_in, const int* in_sizes, int n_in,
                              void* d_out, int out_size, void* d_ws, size_t ws_size,
                              hipStream_t stream) {
    const float* x = (const float*)d_in[0];   // [T, B, 2] float32
    const float* W = (const float*)d_in[1];   // [1, 2]    float32
    float* out     = (float*)d_out;           // [T, B, 1] float32

    const int B = in_sizes[0] / (T_STEPS * 2);   // 262144
    const int grid = B / BLOCK;                  // 1024 blocks, 8 wave32/block

    lif_async_kernel<<<grid, BLOCK, 0, stream>>>(x, W, out, B);
}